// SNNCategoricalPolicy_1168231105207
// MI455X (gfx1250) — compile-verified
//
#include <hip/hip_runtime.h>

// Problem dims
#define BATCH   1024
#define CIN     8
#define HID     300
#define NOUT    2
#define TDIM    300

#define HID_PAD 304              // 19 tiles of 16
#define NTILE   19
#define TCHUNK  16
#define NCHUNK  19               // ceil(300/16)
#define U1S     305              // LDS stride (odd vs 64 banks -> conflict free)
#define NTHREADS 320             // 10 waves

typedef __attribute__((ext_vector_type(2))) float v2f;
typedef __attribute__((ext_vector_type(8))) float v8f;

__device__ __forceinline__ float hash_uniform(unsigned x) {
    x ^= x >> 16; x *= 0x7feb352dU;
    x ^= x >> 15; x *= 0x846ca68bU;
    x ^= x >> 16;
    return ((float)((x >> 8) & 0xFFFFFF) + 0.5f) * (1.0f / 16777216.0f);
}

__launch_bounds__(NTHREADS)
__global__ void snn_policy_kernel(const float* __restrict__ xin,   // [B, CIN, T]
                                  const float* __restrict__ W1,    // [HID, CIN]
                                  const float* __restrict__ W2,    // [NOUT, HID]
                                  float* __restrict__ out)         // logits[B,2] | pi[B] | logp[B]
{
    __shared__ float lds_x [CIN * TDIM];          // 2400 floats: X[c][t] for this batch elem
    __shared__ float lds_u1[TCHUNK * U1S];        // 4880 floats: [tloc][h] currents, then spikes
    __shared__ float lds_w2[NOUT * HID];          // 600
    __shared__ float lds_part[10 * 32];           // per-wave layer2 partials
    __shared__ float lds_i2[32];                  // layer2 currents: [o*16 + tloc]

    const int b    = blockIdx.x;
    const int tid  = threadIdx.x;
    const int lane = tid & 31;
    const int wave = tid >> 5;
    const int kh   = lane >> 4;    // K-half selector for WMMA A/B frags
    const int ln   = lane & 15;    // M (A rows) / N (B,D cols)

    // ---- preload input spikes + W2 into LDS ----
    for (int i = tid; i < CIN * TDIM; i += NTHREADS) lds_x[i]  = xin[b * CIN * TDIM + i];
    for (int i = tid; i < NOUT * HID; i += NTHREADS) lds_w2[i] = W2[i];

    // ---- per-wave WMMA A fragments from W1 (constant across chunks) ----
    // A is 16x4 f32: v0 = K(2*kh), v1 = K(2*kh+1), rows M = ln
    const int tiles[2] = { wave, wave + 10 };
    v2f afrag[2][2];
    for (int j = 0; j < 2; ++j) {
        afrag[j][0] = {}; afrag[j][1] = {};
        if (tiles[j] < NTILE) {
            int h = tiles[j] * 16 + ln;
            if (h < HID) {
                for (int kb = 0; kb < 2; ++kb) {
                    afrag[j][kb].x = W1[h * CIN + kb * 4 + 2 * kh + 0];
                    afrag[j][kb].y = W1[h * CIN + kb * 4 + 2 * kh + 1];
                }
            }
        }
    }

    // IIR constants: eps(n) = n * A * a^n  (alpha kernel, exact, no truncation)
    const float aS = 0.9048374180f;   // exp(-Ts/tau_sr)  = exp(-0.1)
    const float AS = 0.2718281828f;   // e*Ts/tau_sr
    const float aR = 0.6065306597f;   // exp(-Ts/tau_ref) = exp(-0.5)
    const float AR = -27.1828182846f; // -scaleRef*theta*e*Ts/tau_ref
    const float TH = 10.0f;

    // layer-1 per-neuron states (thread h = tid < 300)
    float px = 0.f, py = 0.f, rx = 0.f, ry = 0.f;
    // layer-2 states + spike count (threads 0,1)
    float p2x = 0.f, p2y = 0.f, r2x = 0.f, r2y = 0.f, cnt = 0.f;

    __syncthreads();

    for (int ch = 0; ch < NCHUNK; ++ch) {
        const int t0 = ch * TCHUNK;

        // ---- WMMA: U1[h, t0..t0+15] = W1 x X  (f32 16x16x4, chained K) ----
        v2f bfrag[2];
        for (int kb = 0; kb < 2; ++kb) {
            int t = t0 + ln;
            float b0 = 0.f, b1 = 0.f;
            if (t < TDIM) {
                b0 = lds_x[(kb * 4 + 2 * kh + 0) * TDIM + t];
                b1 = lds_x[(kb * 4 + 2 * kh + 1) * TDIM + t];
            }
            bfrag[kb].x = b0; bfrag[kb].y = b1;
        }
        for (int j = 0; j < 2; ++j) {
            if (tiles[j] < NTILE) {          // wave-uniform: EXEC stays all-ones for WMMA
                v8f acc = {};
                acc = __builtin_amdgcn_wmma_f32_16x16x4_f32(
                          false, afrag[j][0], false, bfrag[0], (short)0, acc, false, false);
                acc = __builtin_amdgcn_wmma_f32_16x16x4_f32(
                          false, afrag[j][1], false, bfrag[1], (short)0, acc, false, false);
                #pragma unroll
                for (int r = 0; r < 8; ++r) {
                    int hM = tiles[j] * 16 + r + 8 * kh;     // D row
                    lds_u1[ln * U1S + hM] = acc[r];          // [tloc][h]
                }
            }
        }
        __syncthreads();   // (1) currents staged

        // ---- layer-1 fused PSP-IIR + integrate-and-fire scan (16 steps) ----
        if (tid < HID) {
            #pragma unroll
            for (int tt = 0; tt < TCHUNK; ++tt) {
                if (t0 + tt >= TDIM) break;
                float i1 = lds_u1[tt * U1S + tid];
                float s  = (px + rx >= TH) ? 1.0f : 0.0f;   // eps(0)=0: i1 enters next step
                float y2 = py + AS * i1;                    // PSP alpha-kernel IIR
                px = aS * (px + y2);  py = aS * y2;
                float yr = ry + AR * s;                     // refractory alpha IIR
                rx = aR * (rx + yr);  ry = aR * yr;
                lds_u1[tt * U1S + tid] = s;                 // leave spikes in place
            }
        }
        __syncthreads();   // (2) s1 spikes ready

        // ---- layer-2 currents: I2[o,tt] = sum_h W2[o,h] * s1[h,tt] ----
        {
            int p = tid & 31, o = p >> 4, tt = p & 15, strip = tid >> 5;
            float acc2 = 0.f;
            if (t0 + tt < TDIM) {
                int h0 = strip * 30;
                #pragma unroll 6
                for (int j2 = 0; j2 < 30; ++j2)
                    acc2 += lds_w2[o * HID + h0 + j2] * lds_u1[tt * U1S + h0 + j2];
            }
            lds_part[strip * 32 + p] = acc2;
        }
        __syncthreads();   // (3)
        if (tid < 32) {
            float v = 0.f;
            #pragma unroll
            for (int k2 = 0; k2 < 10; ++k2) v += lds_part[k2 * 32 + tid];
            lds_i2[tid] = v;
        }
        __syncthreads();   // (4)

        // ---- layer-2 scan + spike counting (threads 0,1 hold persistent state) ----
        if (tid < NOUT) {
            #pragma unroll
            for (int tt = 0; tt < TCHUNK; ++tt) {
                if (t0 + tt >= TDIM) break;
                float i2 = lds_i2[tid * 16 + tt];
                float s  = (p2x + r2x >= TH) ? 1.0f : 0.0f;
                float y2 = p2y + AS * i2;
                p2x = aS * (p2x + y2);  p2y = aS * y2;
                float yr = r2y + AR * s;
                r2x = aR * (r2x + yr);  r2y = aR * yr;
                cnt += s;
            }
        }
        __syncthreads();   // (5) protect LDS before next chunk's WMMA stores
    }

    // ---- finalize: logits, Gumbel-max sample, log-softmax ----
    if (tid < NOUT) lds_i2[tid] = cnt;
    __syncthreads();
    if (tid == 0) {
        float l0 = lds_i2[0], l1 = lds_i2[1];
        out[b * 2 + 0] = l0;
        out[b * 2 + 1] = l1;
        // deterministic Gumbel stand-in for jax.random.gumbel(key 42)
        float g0 = -__logf(-__logf(hash_uniform(0x2a000000u ^ (unsigned)(b * 2 + 0))));
        float g1 = -__logf(-__logf(hash_uniform(0x2a000000u ^ (unsigned)(b * 2 + 1))));
        int pi = (l1 + g1 > l0 + g0) ? 1 : 0;
        float m   = fmaxf(l0, l1);
        float lse = m + __logf(__expf(l0 - m) + __expf(l1 - m));
        out[2 * BATCH + b] = (float)pi;
        out[3 * BATCH + b] = (pi ? l1 : l0) - lse;
    }
}

extern "C" void kernel_launch(void* const* d_in, const int* in_sizes, int n_in,
                              void* d_out, int out_size, void* d_ws, size_t ws_size,
                              hipStream_t stream) {
    const float* xin = (const float*)d_in[0];   // encoded_input [1024,8,300]
    const float* W1  = (const float*)d_in[1];   // [300,8]
    const float* W2  = (const float*)d_in[2];   // [2,300]
    float* out = (float*)d_out;
    snn_policy_kernel<<<dim3(BATCH), dim3(NTHREADS), 0, stream>>>(xin, W1, W2, out);
}